// ResonanceLayer_16286515986757
// MI455X (gfx1250) — compile-verified
//
#include <hip/hip_runtime.h>
#include <math.h>

// ---------------------------------------------------------------------------
// ResonanceLayer on MI455X (gfx1250, wave32).
//
// Per-step tensor reductions are rewritten via sin/cos angle-difference
// identities into 4 GEMMs (u@C^T, v@C^T, u@M^T, v@M^T), computed with
// V_WMMA_F32_16X16X4_F32 (f32 in / f32 acc -> matches f32 reference).
// All 10 steps are fused in one kernel: batch rows are independent, so each
// workgroup owns 16 batch rows, keeps phase/amp/freq in registers (lane
// ownership == WMMA D-matrix layout, stable across steps), and only u/v
// (amp*sin, amp*cos) pass through LDS (64 KB, XOR-swizzled rows to avoid
// 512-stride bank conflicts).
//
// Step kernel uses 1024-thread blocks (32 waves -> 8 waves/SIMD32): the inner
// loop is L2-latency bound (scattered b64 loads of C/M rows), so max
// per-SIMD wave occupancy is the main latency-hiding lever.
// ---------------------------------------------------------------------------

typedef __attribute__((ext_vector_type(2))) float v2f;
typedef __attribute__((ext_vector_type(8))) float v8f;

#define B_SZ 128
#define D_SZ 1024
#define R_SZ 512
#define NSTEPS 10
#define DT_C 0.01f
#define GAMMA_C 0.01f
#define EPS_C 1e-6f
#define AMP_MAX_C 10.0f
#define TWO_PI_F 6.283185307179586f
#define CPL_SCALE 0.04419417382415922f /* 1/sqrt(512) */

// ---------------------------------------------------------------------------
// Kernel 1: proj = x @ W^T   (128x1024) x (512x1024)^T -> (128,512)
// One wave per 16x16 output tile; K-loop in chunks of 4 (f32 WMMA).
// A lane l: M = l%16, K = 2*(l/16)+{0,1}  -> one b64 load from row-major x.
// B lane l: N = l%16, K = 2*(l/16)+{0,1}  -> one b64 load from row-major W.
// ---------------------------------------------------------------------------
__global__ __launch_bounds__(256) void proj_gemm_kernel(
    const float* __restrict__ x, const float* __restrict__ W,
    float* __restrict__ proj) {
  const int lane = threadIdx.x & 31;
  const int wave = threadIdx.x >> 5;
  const int tile = blockIdx.x * 8 + wave;   // 256 tiles total
  const int br = tile >> 5;                 // 0..7   (batch row tile)
  const int cr = tile & 31;                 // 0..31  (R col tile)
  const int nsub = lane & 15;
  const int koff = (lane >> 4) << 1;        // 0 or 2

  const float* xp = x + (br * 16 + nsub) * D_SZ + koff;
  const float* wp = W + (cr * 16 + nsub) * D_SZ + koff;

  v8f acc = {};
  for (int kc = 0; kc < D_SZ / 4; ++kc) {
    v2f a = *(const v2f*)(xp + kc * 4);
    v2f b = *(const v2f*)(wp + kc * 4);
    acc = __builtin_amdgcn_wmma_f32_16x16x4_f32(false, a, false, b,
                                                (short)0, acc, false, false);
  }
  const int rbase = br * 16 + ((lane >> 4) << 3);
  const int col = cr * 16 + nsub;
#pragma unroll
  for (int e = 0; e < 8; ++e) proj[(rbase + e) * R_SZ + col] = acc[e];
}

// ---------------------------------------------------------------------------
// Kernel 2: initial state. One block per batch row.
// rfft(512) done as brute-force DFT with an LDS twiddle table
// (exp(-2pi i m/512)); head bins i<=256, tail i>=257 is the arctan2 path.
// ---------------------------------------------------------------------------
__global__ __launch_bounds__(256) void init_state_kernel(
    const float* __restrict__ proj, const float* __restrict__ basef,
    float* __restrict__ phase, float* __restrict__ amp,
    float* __restrict__ freq) {
  __shared__ float prow[R_SZ];
  __shared__ v2f tw[R_SZ];
  const int b = blockIdx.x;

  for (int i = threadIdx.x; i < R_SZ; i += 256) {
    prow[i] = proj[b * R_SZ + i];
    float ang = -TWO_PI_F * (float)i / (float)R_SZ;
    float s, c;
    sincosf(ang, &s, &c);
    v2f w; w.x = c; w.y = s;          // exp(-2pi i m / 512)
    tw[i] = w;
  }
  __syncthreads();

  for (int i = threadIdx.x; i < R_SZ; i += 256) {
    float ph, am;
    if (i <= R_SZ / 2) {              // rfft head bin
      float re = 0.f, im = 0.f;
      for (int n = 0; n < R_SZ; ++n) {
        v2f w = tw[(i * n) & (R_SZ - 1)];
        re = fmaf(prow[n], w.x, re);
        im = fmaf(prow[n], w.y, im);
      }
      ph = atan2f(im, re);
      am = sqrtf(re * re + im * im) + EPS_C;
    } else {                          // tail
      float t = prow[i];
      float a = fabsf(t) + EPS_C;
      ph = atan2f(t, a);
      am = a;
    }
    am = fminf(fmaxf(am, EPS_C), AMP_MAX_C);
    ph = fmodf(ph, TWO_PI_F);
    if (ph < 0.f) ph += TWO_PI_F;
    phase[b * R_SZ + i] = ph;
    amp[b * R_SZ + i] = am;
    freq[b * R_SZ + i] = basef[i];
  }
}

// ---------------------------------------------------------------------------
// Kernel 3: fused 10-step dynamics. grid = B/16 blocks x 1024 threads
// (32 waves = 8 waves/SIMD32). Wave w owns i-tile w. Lane l owns state for
//   (b = e + 8*(l/16), i = w*16 + l%16), e = 0..7   -- the WMMA D layout,
// stable across steps, so phase/amp/freq live in registers the whole time.
// Per step: write u/v to LDS (swizzled), barrier, 4 WMMA accumulations over
// j (128 chunks of K=4), then the elementwise update in registers.
// ---------------------------------------------------------------------------
__global__ __launch_bounds__(1024) void resonance_steps_kernel(
    const float* __restrict__ coupling, const float* __restrict__ modulation,
    const float* __restrict__ decay, const float* __restrict__ phase_in,
    const float* __restrict__ amp_in, const float* __restrict__ freq_in,
    float* __restrict__ amp_out) {
  __shared__ float u_lds[16 * R_SZ];  // 32 KB
  __shared__ float v_lds[16 * R_SZ];  // 32 KB
  const int lane = threadIdx.x & 31;
  const int wave = threadIdx.x >> 5;  // 0..31 -> i-tile id
  const int half = lane >> 4;         // 0/1
  const int nsub = lane & 15;
  const int b0 = blockIdx.x * 16;
  const int i_own = wave * 16 + nsub; // this lane's oscillator column

  float phase_r[8], amp_r[8], freq_r[8];
  const float decay_r = decay[i_own];
#pragma unroll
  for (int e = 0; e < 8; ++e) {
    const int b = b0 + e + half * 8;
    phase_r[e] = phase_in[b * R_SZ + i_own];
    amp_r[e] = amp_in[b * R_SZ + i_own];
    freq_r[e] = freq_in[b * R_SZ + i_own];
  }

  const int koff = half << 1;  // 0 or 2
  const float* cp = coupling + i_own * R_SZ + koff;
  const float* mp = modulation + i_own * R_SZ + koff;

  for (int step = 0; step < NSTEPS; ++step) {
    // --- u = amp*sin(phase), v = amp*cos(phase), XOR-swizzled per row ---
#pragma unroll
    for (int e = 0; e < 8; ++e) {
      const int brow = e + half * 8;  // 0..15 within batch tile
      float s, c;
      sincosf(phase_r[e], &s, &c);
      const int jx = i_own ^ (brow << 2);
      u_lds[brow * R_SZ + jx] = amp_r[e] * s;
      v_lds[brow * R_SZ + jx] = amp_r[e] * c;
    }
    __syncthreads();

    // --- 4 GEMM accumulations over j ---
    v8f su = {}, sv = {}, tu = {}, tv = {};
    for (int jc = 0; jc < R_SZ / 4; ++jc) {
      const int j = jc * 4 + koff;
      const int jx = j ^ (nsub << 2);  // A row M = nsub, swizzled
      v2f au = *(const v2f*)&u_lds[nsub * R_SZ + jx];
      v2f av = *(const v2f*)&v_lds[nsub * R_SZ + jx];
      v2f bc = *(const v2f*)(cp + jc * 4);
      v2f bm = *(const v2f*)(mp + jc * 4);
      su = __builtin_amdgcn_wmma_f32_16x16x4_f32(false, au, false, bc,
                                                 (short)0, su, false, false);
      sv = __builtin_amdgcn_wmma_f32_16x16x4_f32(false, av, false, bc,
                                                 (short)0, sv, false, false);
      tu = __builtin_amdgcn_wmma_f32_16x16x4_f32(false, au, false, bm,
                                                 (short)0, tu, false, false);
      tv = __builtin_amdgcn_wmma_f32_16x16x4_f32(false, av, false, bm,
                                                 (short)0, tv, false, false);
    }

    // --- elementwise update in registers ---
#pragma unroll
    for (int e = 0; e < 8; ++e) {
      float s, c;
      sincosf(phase_r[e], &s, &c);
      const float Su = su[e] * CPL_SCALE;
      const float Sv = sv[e] * CPL_SCALE;
      const float phase_up = c * Su - s * Sv;
      const float amp_cpl = c * Sv + s * Su;
      const float freq_up = c * tu[e] - s * tv[e];
      float np = phase_r[e] + DT_C * (freq_r[e] + phase_up);
      np = fmodf(np, TWO_PI_F);
      if (np < 0.f) np += TWO_PI_F;
      float na = amp_r[e] + DT_C * (-decay_r * amp_r[e] + amp_cpl);
      na = fminf(fmaxf(na, EPS_C), AMP_MAX_C);
      phase_r[e] = np;
      amp_r[e] = na;
      freq_r[e] += DT_C * GAMMA_C * freq_up;
    }
    __syncthreads();  // protect u/v before next step overwrites
  }

  // --- final amp out ---
#pragma unroll
  for (int e = 0; e < 8; ++e) {
    const int b = b0 + e + half * 8;
    amp_out[b * R_SZ + i_own] = amp_r[e];
  }
}

// ---------------------------------------------------------------------------
extern "C" void kernel_launch(void* const* d_in, const int* in_sizes, int n_in,
                              void* d_out, int out_size, void* d_ws,
                              size_t ws_size, hipStream_t stream) {
  const float* x = (const float*)d_in[0];           // (128,1024)
  const float* coupling = (const float*)d_in[1];    // (512,512), zero diag
  const float* decay = (const float*)d_in[2];       // (512,)
  const float* W = (const float*)d_in[3];           // (512,1024)
  const float* modulation = (const float*)d_in[4];  // (512,512)
  const float* basef = (const float*)d_in[5];       // (512,)

  float* ws = (float*)d_ws;
  float* proj = ws;                       // 128*512
  float* phase = ws + B_SZ * R_SZ;        // 128*512
  float* amp = ws + 2 * B_SZ * R_SZ;      // 128*512
  float* freq = ws + 3 * B_SZ * R_SZ;     // 128*512

  proj_gemm_kernel<<<32, 256, 0, stream>>>(x, W, proj);
  init_state_kernel<<<B_SZ, 256, 0, stream>>>(proj, basef, phase, amp, freq);
  resonance_steps_kernel<<<B_SZ / 16, 1024, 0, stream>>>(
      coupling, modulation, decay, phase, amp, freq, (float*)d_out);
}